// TemporalGraphNetworkGAT_EdgeRelation_46205258170460
// MI455X (gfx1250) — compile-verified
//
#include <hip/hip_runtime.h>
#include <hip/hip_bf16.h>

// -------- CDNA5 WMMA types --------
typedef __attribute__((ext_vector_type(2))) float v2f;
typedef __attribute__((ext_vector_type(8))) float v8f;

#define THREADS 256

// ---------------------------------------------------------------------------
// Utility kernels
// ---------------------------------------------------------------------------
__global__ void fill_kernel(float* __restrict__ p, float v, int n) {
    int t = blockIdx.x * blockDim.x + threadIdx.x;
    if (t < n) p[t] = v;
}

__device__ __forceinline__ float leaky02(float x) {
    return x > 0.0f ? x : 0.2f * x;
}

// float atomic max via sign-split integer ordering trick
__device__ __forceinline__ void atomicMaxFloat(float* addr, float v) {
    if (v >= 0.0f)
        atomicMax((int*)addr, __float_as_int(v));
    else
        atomicMin((unsigned int*)addr, __float_as_uint(v));
}

// virtual edge list with self loops appended: e < E -> (src[e], dst[e]); else (i,i)
__device__ __forceinline__ void edge_sd(const int* __restrict__ src,
                                        const int* __restrict__ dst,
                                        int e, int E, int& s, int& d) {
    if (e < E) { s = src[e]; d = dst[e]; }
    else       { s = e - E;  d = e - E;  }
}

// ---------------------------------------------------------------------------
// Dense GEMM via V_WMMA_F32_16X16X4_F32 : D[M,Nc] = A[M,K] @ B[K,Nc]
// One wave computes one 16x16 output tile. Nc multiple of 16.
// ---------------------------------------------------------------------------
__global__ void wmma_gemm_f32(const float* __restrict__ A,
                              const float* __restrict__ B,
                              float* __restrict__ D,
                              int M, int K, int Nc) {
    const int lane = threadIdx.x & 31;
    const int half = lane >> 4;       // 0: lanes 0-15, 1: lanes 16-31
    const int lm   = lane & 15;
    const int wavesPerBlock = blockDim.x >> 5;
    const int tile = blockIdx.x * wavesPerBlock + (threadIdx.x >> 5);

    const int colTiles   = Nc >> 4;
    const int rowTiles   = (M + 15) >> 4;
    const int totalTiles = rowTiles * colTiles;
    if (tile >= totalTiles) return;           // wave-uniform: EXEC stays full

    const int rowTile = tile / colTiles;
    const int colTile = tile - rowTile * colTiles;

    // A fragment row for this lane (clamped, stores are guarded)
    const int arow = min(rowTile * 16 + lm, M - 1);
    const int nb   = colTile * 16 + lm;       // B/D column for this lane

    const float* __restrict__ arp = A + (size_t)arow * K;

    v8f acc = {};
    for (int k = 0; k < K; k += 4) {
        const int kk = k + (half << 1);       // half-wave handles K=kk, kk+1
        float2 av = *(const float2*)(arp + kk);
        v2f a; a[0] = av.x; a[1] = av.y;
        v2f b;
        b[0] = B[(size_t)kk * Nc + nb];
        b[1] = B[(size_t)(kk + 1) * Nc + nb];
        acc = __builtin_amdgcn_wmma_f32_16x16x4_f32(
            false, a, false, b, (short)0, acc, false, false);
    }

    // D tile: VGPR r -> row r + 8*half, col = lm
    const int mbase = rowTile * 16 + (half << 3);
    #pragma unroll
    for (int r = 0; r < 8; ++r) {
        const int m = mbase + r;
        if (m < M) D[(size_t)m * Nc + nb] = acc[r];
    }
}

// ---------------------------------------------------------------------------
// alpha_s[n,h] = sum_c h_lin[n,h,c]*a_src[h,c] ; same for alpha_d
// one thread per (node, head)
// ---------------------------------------------------------------------------
__global__ void alpha_kernel(const float* __restrict__ h_lin,
                             const float* __restrict__ a_src,
                             const float* __restrict__ a_dst,
                             float* __restrict__ as, float* __restrict__ ad,
                             int N) {
    int t = blockIdx.x * blockDim.x + threadIdx.x;
    if (t >= N * 4) return;
    int n = t >> 2, h = t & 3;
    const float* __restrict__ hp = h_lin + (size_t)n * 256 + h * 64;
    const float* __restrict__ sp = a_src + h * 64;
    const float* __restrict__ dp = a_dst + h * 64;
    float sa = 0.0f, sd = 0.0f;
    #pragma unroll 8
    for (int c = 0; c < 64; ++c) {
        float hv = hp[c];
        sa += hv * sp[c];
        sd += hv * dp[c];
    }
    as[t] = sa;
    ad[t] = sd;
}

// ---------------------------------------------------------------------------
// Pass A: segment max of leaky-relu logits over dst
// ---------------------------------------------------------------------------
__global__ void edge_max_kernel(const int* __restrict__ src,
                                const int* __restrict__ dst,
                                const float* __restrict__ as,
                                const float* __restrict__ ad,
                                float* __restrict__ mmax,
                                int E, int N) {
    int e = blockIdx.x * blockDim.x + threadIdx.x;
    if (e >= E + N) return;
    int s, d; edge_sd(src, dst, e, E, s, d);
    #pragma unroll
    for (int h = 0; h < 4; ++h) {
        float lv = leaky02(as[s * 4 + h] + ad[d * 4 + h]);
        atomicMaxFloat(&mmax[d * 4 + h], lv);
    }
}

// ---------------------------------------------------------------------------
// Pass B: ex = exp(e - m[dst]) stored per edge; segment sum over dst
// ---------------------------------------------------------------------------
__global__ void edge_expsum_kernel(const int* __restrict__ src,
                                   const int* __restrict__ dst,
                                   const float* __restrict__ as,
                                   const float* __restrict__ ad,
                                   const float* __restrict__ mmax,
                                   float* __restrict__ ssum,
                                   float* __restrict__ exb,
                                   int E, int N) {
    int e = blockIdx.x * blockDim.x + threadIdx.x;
    if (e >= E + N) return;
    int s, d; edge_sd(src, dst, e, E, s, d);
    #pragma unroll
    for (int h = 0; h < 4; ++h) {
        float lv = leaky02(as[s * 4 + h] + ad[d * 4 + h]);
        float ex = __expf(lv - mmax[d * 4 + h]);
        exb[(size_t)e * 4 + h] = ex;
        atomicAdd(&ssum[d * 4 + h], ex);
    }
}

// ---------------------------------------------------------------------------
// Pass C: out[dst,h,c] += h_lin[src,h,c] * (ex/s[dst,h]); thread per (edge,c)
// ---------------------------------------------------------------------------
__global__ void edge_aggr_kernel(const int* __restrict__ src,
                                 const int* __restrict__ dst,
                                 const float* __restrict__ h_lin,
                                 const float* __restrict__ exb,
                                 const float* __restrict__ ssum,
                                 float* __restrict__ acc,
                                 int E, int N) {
    long long t = (long long)blockIdx.x * blockDim.x + threadIdx.x;
    long long total = (long long)(E + N) * 64;
    if (t >= total) return;
    int e = (int)(t >> 6);
    int c = (int)(t & 63);
    int s, d; edge_sd(src, dst, e, E, s, d);
    const float* __restrict__ hs = h_lin + (size_t)s * 256;
    float* __restrict__ od = acc + (size_t)d * 256;
    #pragma unroll
    for (int h = 0; h < 4; ++h) {
        float coeff = exb[(size_t)e * 4 + h] / ssum[d * 4 + h];
        atomicAdd(&od[h * 64 + c], hs[h * 64 + c] * coeff);
    }
}

// ---------------------------------------------------------------------------
// head mean + bias + relu : node_out[n,c] = relu(mean_h acc[n,h,c] + bias[c])
// ---------------------------------------------------------------------------
__global__ void node_finalize_kernel(const float* __restrict__ acc,
                                     const float* __restrict__ bias,
                                     float* __restrict__ node_out, int N) {
    int t = blockIdx.x * blockDim.x + threadIdx.x;
    if (t >= N * 64) return;
    int n = t >> 6, c = t & 63;
    const float* __restrict__ ap = acc + (size_t)n * 256 + c;
    float v = 0.25f * (ap[0] + ap[64] + ap[128] + ap[192]) + bias[c];
    node_out[t] = v > 0.0f ? v : 0.0f;
}

// ---------------------------------------------------------------------------
// Relation head: out[E,R] = concat(h[src],h[dst]) @ Wr + br via WMMA f32
// one wave = one 16-edge tile; K = 128; only R(=9) columns valid.
// ---------------------------------------------------------------------------
__global__ void rel_wmma_kernel(const int* __restrict__ src,
                                const int* __restrict__ dst,
                                const float* __restrict__ hfin,
                                const float* __restrict__ Wr,
                                const float* __restrict__ br,
                                float* __restrict__ out,
                                int E, int R) {
    const int lane = threadIdx.x & 31;
    const int half = lane >> 4;
    const int lm   = lane & 15;
    const int wavesPerBlock = blockDim.x >> 5;
    const int etile = blockIdx.x * wavesPerBlock + (threadIdx.x >> 5);
    const int tiles = (E + 15) >> 4;
    if (etile >= tiles) return;               // wave-uniform

    const int erow  = etile * 16 + lm;
    const int ecl   = min(erow, E - 1);
    const float* __restrict__ fs = hfin + (size_t)src[ecl] * 64;
    const float* __restrict__ fd = hfin + (size_t)dst[ecl] * 64;
    const bool colOK = lm < R;

    v8f acc = {};
    for (int k = 0; k < 128; k += 4) {
        const int kk = k + (half << 1);       // even; pair stays on one side of 64
        const float* base = (kk < 64) ? (fs + kk) : (fd + (kk - 64));
        v2f a; a[0] = base[0]; a[1] = base[1];
        v2f b;
        b[0] = colOK ? Wr[(size_t)kk * R + lm] : 0.0f;
        b[1] = colOK ? Wr[(size_t)(kk + 1) * R + lm] : 0.0f;
        acc = __builtin_amdgcn_wmma_f32_16x16x4_f32(
            false, a, false, b, (short)0, acc, false, false);
    }

    const int mbase = etile * 16 + (half << 3);
    const float bb = colOK ? br[lm] : 0.0f;
    #pragma unroll
    for (int r = 0; r < 8; ++r) {
        const int e = mbase + r;
        if (e < E && colOK) out[(size_t)e * R + lm] = acc[r] + bb;
    }
}

// ---------------------------------------------------------------------------
// Host-side orchestration
// ---------------------------------------------------------------------------
static inline int cdiv(long long a, long long b) { return (int)((a + b - 1) / b); }

extern "C" void kernel_launch(void* const* d_in, const int* in_sizes, int n_in,
                              void* d_out, int out_size, void* d_ws, size_t ws_size,
                              hipStream_t stream) {
    const float* x     = (const float*)d_in[0];
    const int*   eidx  = (const int*)  d_in[1];
    const float* W1    = (const float*)d_in[2];
    const float* as1   = (const float*)d_in[3];
    const float* ad1   = (const float*)d_in[4];
    const float* b1    = (const float*)d_in[5];
    const float* W2    = (const float*)d_in[6];
    const float* as2   = (const float*)d_in[7];
    const float* ad2   = (const float*)d_in[8];
    const float* b2    = (const float*)d_in[9];
    const float* Wr    = (const float*)d_in[10];
    const float* br    = (const float*)d_in[11];
    float* out = (float*)d_out;

    const int F = 128;
    const int N = in_sizes[0] / F;        // 50000
    const int E = in_sizes[1] / 2;        // 800000
    const int R = in_sizes[11];           // 9
    const int Etot = E + N;               // with self loops

    const int* src = eidx;
    const int* dst = eidx + E;

    // workspace layout (floats)
    float* ws = (float*)d_ws;
    size_t off = 0;
    float* h_lin = ws + off; off += (size_t)N * 256;   // linear output [N,H,C]
    float* accb  = ws + off; off += (size_t)N * 256;   // segment-sum accumulator
    float* node1 = ws + off; off += (size_t)N * 64;    // layer-1 node output
    float* node2 = ws + off; off += (size_t)N * 64;    // layer-2 node output
    float* alS   = ws + off; off += (size_t)N * 4;
    float* alD   = ws + off; off += (size_t)N * 4;
    float* mmax  = ws + off; off += (size_t)N * 4;
    float* ssum  = ws + off; off += (size_t)N * 4;
    float* exb   = ws + off; off += (size_t)Etot * 4;
    (void)ws_size; (void)n_in;

    const int gemmTiles = ((N + 15) / 16) * (256 / 16);
    const int gemmBlocks = cdiv(gemmTiles, THREADS / 32);

    for (int layer = 0; layer < 2; ++layer) {
        const float* A      = layer == 0 ? x   : node1;
        const float* W      = layer == 0 ? W1  : W2;
        const float* a_src  = layer == 0 ? as1 : as2;
        const float* a_dst  = layer == 0 ? ad1 : ad2;
        const float* bias   = layer == 0 ? b1  : b2;
        float*       nodeO  = layer == 0 ? node1 : node2;
        const int    K      = layer == 0 ? F : 64;

        // init accumulators
        fill_kernel<<<cdiv((long long)N * 256, THREADS), THREADS, 0, stream>>>(accb, 0.0f, N * 256);
        fill_kernel<<<cdiv((long long)N * 4, THREADS), THREADS, 0, stream>>>(mmax, -1.0e30f, N * 4);
        fill_kernel<<<cdiv((long long)N * 4, THREADS), THREADS, 0, stream>>>(ssum, 0.0f, N * 4);

        // h = A @ W   (fp32 WMMA)
        wmma_gemm_f32<<<gemmBlocks, THREADS, 0, stream>>>(A, W, h_lin, N, K, 256);

        // attention coefficients
        alpha_kernel<<<cdiv((long long)N * 4, THREADS), THREADS, 0, stream>>>(
            h_lin, a_src, a_dst, alS, alD, N);

        // softmax over destination segments
        edge_max_kernel<<<cdiv(Etot, THREADS), THREADS, 0, stream>>>(
            src, dst, alS, alD, mmax, E, N);
        edge_expsum_kernel<<<cdiv(Etot, THREADS), THREADS, 0, stream>>>(
            src, dst, alS, alD, mmax, ssum, exb, E, N);

        // weighted message aggregation
        edge_aggr_kernel<<<cdiv((long long)Etot * 64, THREADS), THREADS, 0, stream>>>(
            src, dst, h_lin, exb, ssum, accb, E, N);

        // head mean + bias + relu
        node_finalize_kernel<<<cdiv((long long)N * 64, THREADS), THREADS, 0, stream>>>(
            accb, bias, nodeO, N);
    }

    // relation head over original edges (no self loops)
    const int relTiles = (E + 15) / 16;
    rel_wmma_kernel<<<cdiv(relTiles, THREADS / 32), THREADS, 0, stream>>>(
        src, dst, node2, Wr, br, out, E, R);
    (void)out_size;
}